// RNN_3_76965813944824
// MI455X (gfx1250) — compile-verified
//
#include <hip/hip_runtime.h>
#include <hip/hip_bf16.h>

typedef float v2f __attribute__((ext_vector_type(2)));
typedef float v8f __attribute__((ext_vector_type(8)));

#define HID   10
#define PADH  16
#define PIPE  32

// ---------------------------------------------------------------------------
// Branchless fast tanh: no EXEC divergence, short dependency chain.
//   tanh(y) = 1 - 2/(1 + e^{2y})   via native v_exp_f32 + v_rcp_f32 + Newton
//   small-|y| path y - y^3/3 selected with one v_cndmask (no branch).
// ---------------------------------------------------------------------------
__device__ __forceinline__ float tanh_fast(float y) {
    float z = fminf(fmaxf(y, -15.0f), 15.0f);                  // v_med3, keeps exp finite
    float e = __builtin_amdgcn_exp2f(z * 2.8853900817779268f); // 2*log2(e)
    float d = e + 1.0f;
    float r = __builtin_amdgcn_rcpf(d);
    r = r + r * __builtin_fmaf(-d, r, 1.0f);                   // one Newton step
    float hexp = __builtin_fmaf(-2.0f, r, 1.0f);               // 1 - 2r
    float hp   = y - 0.33333333f * y * y * y;                  // |y| small
    return (fabsf(y) < 0.05f) ? hp : hexp;
}

// ---------------------------------------------------------------------------
// Phase A: U[t][j] = b_ih[j] + b_hh[j] + sum_k W_ih[j,k]*x[t,k]  (j < HID)
//          U[t][HID] = b_fc   (feeds the fused output column of B)
//          U[t][j>HID] = 0
// ---------------------------------------------------------------------------
__global__ void rnn_u_kernel(const float* __restrict__ src,
                             const float* __restrict__ W_ih,
                             const float* __restrict__ b_ih,
                             const float* __restrict__ b_hh,
                             const float* __restrict__ b_fc,
                             float* __restrict__ U, int seq) {
    int tid = blockIdx.x * blockDim.x + threadIdx.x;
    if (tid >= seq * PADH) return;
    int j = tid & (PADH - 1);
    int t = tid >> 4;
    float u = 0.0f;
    if (j < HID) {
        const float* x = src + (size_t)3 * t;
        u = b_ih[j] + b_hh[j]
          + W_ih[j * 3 + 0] * x[0]
          + W_ih[j * 3 + 1] * x[1]
          + W_ih[j * 3 + 2] * x[2];
    } else if (j == HID) {
        u = b_fc[0];
    }
    U[tid] = u;
}

// ---------------------------------------------------------------------------
// Phase B: single-wave serial scan, fused output column, independent zero-C WMMAs.
//   B[k][n] = W_hh[n][k] (n<10), W_fc[k] (n==10), 0 otherwise.
//   Step t:  dA=A0*B0, dB=A1*B1, dC=A2*B2  (3 INDEPENDENT v_wmma, C = inline 0,
//            back-to-back in the XDL pipe -> one WMMA latency on the chain)
//            y = (dA[0]+dB[0]) + (dC[0]+u_t(n))
//            lane n<10: h_t[n] = tanh(y_n);  lane 10: y_10 = out[t-1] (readlane, store)
//            A regs rebuilt with v_readlane (h_0..h_9 -> SGPRs) + v_cndmask per half.
//   u feed is register-pipelined PIPE steps ahead, refills always in-bounds.
// ---------------------------------------------------------------------------
template <bool USE_U>
__global__ __launch_bounds__(32)
void rnn_seq_kernel(const float* __restrict__ src,
                    const float* __restrict__ U,
                    const float* __restrict__ W_ih,
                    const float* __restrict__ W_hh,
                    const float* __restrict__ b_ih,
                    const float* __restrict__ b_hh,
                    const float* __restrict__ W_fc,
                    const float* __restrict__ b_fc,
                    float* __restrict__ out, int seq) {
    const int lane = threadIdx.x;       // 0..31, wave32
    const int n    = lane & 15;         // column this lane owns
    const int kh   = (lane >> 4) << 1;  // K sub-offset: 0 (lanes 0-15) / 2 (lanes 16-31)
    const bool lo  = (lane < 16);

    const float bfc = b_fc[0];

    // Fallback-path u coefficients (also carries the b_fc column n==HID).
    float wi0 = 0.f, wi1 = 0.f, wi2 = 0.f, bsum = 0.f;
    if (!USE_U) {
        if (n < HID) {
            wi0 = W_ih[n * 3 + 0];
            wi1 = W_ih[n * 3 + 1];
            wi2 = W_ih[n * 3 + 2];
            bsum = b_ih[n] + b_hh[n];
        } else if (n == HID) {
            bsum = bfc;
        }
    }

    // B = [W_hh^T | W_fc | 0...] zero-padded to 16x16, three 4x16 K-slices.
    v2f B0, B1, B2;
    {
        auto w = [&](int r, int c) -> float {
            if (r < HID && c < HID) return W_hh[r * HID + c];
            if (r == HID && c < HID) return W_fc[c];
            return 0.0f;
        };
        B0.x = w(n, 0 + kh); B0.y = w(n, 1 + kh);
        B1.x = w(n, 4 + kh); B1.y = w(n, 5 + kh);
        B2.x = w(n, 8 + kh); B2.y = w(n, 9 + kh);
    }

    v2f A0 = {0.f, 0.f}, A1 = {0.f, 0.f}, A2 = {0.f, 0.f};      // h_{-1} = 0
    const v8f czero = {0.f, 0.f, 0.f, 0.f, 0.f, 0.f, 0.f, 0.f}; // lowers to inline 0

    // One step: consumes u_t, updates h (in A regs), stores out[t-1] (t>=1).
    auto step = [&](int t, float u) {
        v8f da = __builtin_amdgcn_wmma_f32_16x16x4_f32(false, A0, false, B0, (short)0, czero, false, false);
        v8f db = __builtin_amdgcn_wmma_f32_16x16x4_f32(false, A1, false, B1, (short)0, czero, false, false);
        v8f dc = __builtin_amdgcn_wmma_f32_16x16x4_f32(false, A2, false, B2, (short)0, czero, false, false);
        float y = (da[0] + db[0]) + (dc[0] + u);   // lane n<10: y_n;  lane 10: out[t-1]
        if (t > 0) {
            int oi = __builtin_amdgcn_readlane(__float_as_int(y), HID);
            out[t - 1] = __int_as_float(oi);       // all lanes store same scalar value
        }
        float h = tanh_fast(y);
        int hb = __float_as_int(h);
        // Broadcast h_0..h_9 to SGPRs; rebuild A pairs {h[k],h[k+1]} per wave half.
        int h0 = __builtin_amdgcn_readlane(hb, 0);
        int h1 = __builtin_amdgcn_readlane(hb, 1);
        int h2 = __builtin_amdgcn_readlane(hb, 2);
        int h3 = __builtin_amdgcn_readlane(hb, 3);
        int h4 = __builtin_amdgcn_readlane(hb, 4);
        int h5 = __builtin_amdgcn_readlane(hb, 5);
        int h6 = __builtin_amdgcn_readlane(hb, 6);
        int h7 = __builtin_amdgcn_readlane(hb, 7);
        int h8 = __builtin_amdgcn_readlane(hb, 8);
        int h9 = __builtin_amdgcn_readlane(hb, 9);
        A0.x = __int_as_float(lo ? h0 : h2);
        A0.y = __int_as_float(lo ? h1 : h3);
        A1.x = __int_as_float(lo ? h4 : h6);
        A1.y = __int_as_float(lo ? h5 : h7);
        A2.x = __int_as_float(lo ? h8 : 0);        // K=10 pad
        A2.y = __int_as_float(lo ? h9 : 0);        // K=11 pad
    };

    auto load_u = [&](int t) -> float {    // t must be < seq
        if (USE_U) return U[(size_t)t * PADH + n];
        const float* x = src + (size_t)3 * t;
        return bsum + wi0 * x[0] + wi1 * x[1] + wi2 * x[2];
    };

    int base = 0;
    if (seq >= 2 * PIPE) {
        float ubuf[PIPE];
#pragma unroll
        for (int i = 0; i < PIPE; ++i) ubuf[i] = load_u(i);

        for (; base + 2 * PIPE <= seq; base += PIPE) {
            // All refills in this block are guaranteed in-bounds: no clamps.
            const float* un = USE_U ? (U + (size_t)(base + PIPE) * PADH + n) : nullptr;
            const float* xn = USE_U ? nullptr : (src + (size_t)3 * (base + PIPE));
            {   // Prefetch ~8 blocks ahead into L2 (global_prefetch_b8).
                int pft = base + 8 * PIPE;
                if (pft + PIPE <= seq) {
                    if (USE_U) __builtin_prefetch(U + (size_t)pft * PADH + lane * 16, 0, 1);
                    else       __builtin_prefetch(src + (size_t)pft * 3 + lane * 4, 0, 1);
                }
            }
#pragma unroll
            for (int i = 0; i < PIPE; ++i) {
                float u = ubuf[i];
                ubuf[i] = USE_U ? un[(size_t)i * PADH]
                                : (bsum + wi0 * xn[3 * i + 0]
                                        + wi1 * xn[3 * i + 1]
                                        + wi2 * xn[3 * i + 2]);
                step(base + i, u);
            }
        }
        // Drain the already-buffered block [base, base+PIPE).
#pragma unroll
        for (int i = 0; i < PIPE; ++i) step(base + i, ubuf[i]);
        base += PIPE;
    }
    for (; base < seq; ++base) step(base, load_u(base));  // short direct tail

    // Flush out[seq-1]: one more WMMA pass over the final h.
    if (seq > 0) {
        v8f da = __builtin_amdgcn_wmma_f32_16x16x4_f32(false, A0, false, B0, (short)0, czero, false, false);
        v8f db = __builtin_amdgcn_wmma_f32_16x16x4_f32(false, A1, false, B1, (short)0, czero, false, false);
        v8f dc = __builtin_amdgcn_wmma_f32_16x16x4_f32(false, A2, false, B2, (short)0, czero, false, false);
        float ylast = (da[0] + db[0]) + dc[0];
        int oi = __builtin_amdgcn_readlane(__float_as_int(ylast), HID);
        out[seq - 1] = __int_as_float(oi) + bfc;
    }
}

// ---------------------------------------------------------------------------
extern "C" void kernel_launch(void* const* d_in, const int* in_sizes, int n_in,
                              void* d_out, int out_size, void* d_ws, size_t ws_size,
                              hipStream_t stream) {
    const float* src  = (const float*)d_in[0];
    const float* W_ih = (const float*)d_in[1];
    const float* W_hh = (const float*)d_in[2];
    const float* b_ih = (const float*)d_in[3];
    const float* b_hh = (const float*)d_in[4];
    const float* W_fc = (const float*)d_in[5];
    const float* b_fc = (const float*)d_in[6];
    float* out = (float*)d_out;

    int seq = in_sizes[0] / 3;  // src is (SEQ, 1, 3)
    size_t need = (size_t)seq * PADH * sizeof(float);

    if (ws_size >= need) {
        float* U = (float*)d_ws;
        int total = seq * PADH;
        rnn_u_kernel<<<(total + 255) / 256, 256, 0, stream>>>(src, W_ih, b_ih, b_hh, b_fc, U, seq);
        rnn_seq_kernel<true><<<1, 32, 0, stream>>>(src, U, W_ih, W_hh, b_ih, b_hh,
                                                   W_fc, b_fc, out, seq);
    } else {
        rnn_seq_kernel<false><<<1, 32, 0, stream>>>(src, nullptr, W_ih, W_hh, b_ih, b_hh,
                                                    W_fc, b_fc, out, seq);
    }
}